// TriplaneEncoding_28733331210884
// MI455X (gfx1250) — compile-verified
//
#include <hip/hip_runtime.h>
#include <stdint.h>

// TriplaneEncoding for gfx1250 (MI455X).
// N = 1M points, C = 4 channels, levels r = 128,256,512,1024.
// out[n, l*12 + p*4 + c] = bilinear(triplane_l[p,c], proj_p(x2)) * m[l,p,c] * (1 - |x2[p]|/2)
// where m[l,p,c] = 0.5*(triline_l[p,c,r/2-1,0] + triline_l[p,c,r/2,0])   (degenerate W=1 grid_sample)
//
// Memory-bound: ~268 MB HBM (192 MB output stream + 64 MB L2-resident grids + 12 MB x).
// Gathers (~768 MB) hit L2 (192 MB) after first touch; we minimize L2 *request count*
// by fetching each bilinear row-pair as one b64 instead of two b32s.

typedef float v4f __attribute__((ext_vector_type(4)));
typedef float v2f __attribute__((ext_vector_type(2)));

#define NC 4
#define NLEV 4
#define BLK 256

__global__ __launch_bounds__(BLK)
void triplane_enc_kernel(const float* __restrict__ xin,
                         const float* __restrict__ tp0, const float* __restrict__ tp1,
                         const float* __restrict__ tp2, const float* __restrict__ tp3,
                         const float* __restrict__ tl0, const float* __restrict__ tl1,
                         const float* __restrict__ tl2, const float* __restrict__ tl3,
                         float* __restrict__ out, int N)
{
    __shared__ float s_rows[96];  // [48 entries][2 rows]: triline rows r/2-1 and r/2
    __shared__ float s_m[48];     // reduced line constants m[l*12 + p*4 + c]

    const float* tps[NLEV] = {tp0, tp1, tp2, tp3};
    const float* tls[NLEV] = {tl0, tl1, tl2, tl3};

    const int tid = threadIdx.x;

    // ---- Stage the 96 triline scalars into LDS via async global->LDS DMA ----
    if (tid < 96) {
        const int e    = tid >> 1;        // entry 0..47 = l*12 + p*4 + c
        const int half = tid & 1;         // row r/2-1 or r/2
        const int l    = e / 12;
        const int rem  = e % 12;
        const int p    = rem >> 2;
        const int c    = rem & 3;
        const int r    = 128 << l;
        const float* g = tls[l] + (size_t)(p * NC + c) * r + (r / 2 - 1 + half);
        // Low 32 bits of the generic address of an LDS object are the LDS byte
        // offset (local->flat addrspacecast keeps the offset in addr[31:0]).
        unsigned lds_addr = (unsigned)(size_t)&s_rows[tid];
        unsigned long long gaddr = (unsigned long long)(size_t)g;
        asm volatile("global_load_async_to_lds_b32 %0, %1, off"
                     :: "v"(lds_addr), "v"(gaddr) : "memory");
    }
    asm volatile("s_wait_asynccnt 0" ::: "memory");
    __syncthreads();
    if (tid < 48) {
        s_m[tid] = 0.5f * (s_rows[2 * tid] + s_rows[2 * tid + 1]);
    }
    __syncthreads();

    const int n = blockIdx.x * BLK + tid;
    if (n >= N) return;

    // ---- Load point, map [0,1] -> [-1,1] ----
    const float xc0 = xin[3 * n + 0] * 2.0f - 1.0f;
    const float xc1 = xin[3 * n + 1] * 2.0f - 1.0f;
    const float xc2 = xin[3 * n + 2] * 2.0f - 1.0f;

    // Line weights: w = 1 - |x2|/2 (per plane p, line dim = p)
    float wl[3];
    wl[0] = 1.0f - 0.5f * fabsf(xc0);
    wl[1] = 1.0f - 0.5f * fabsf(xc1);
    wl[2] = 1.0f - 0.5f * fabsf(xc2);

    // Plane projections: p0 -> (x, z), p1 -> (y, x), p2 -> (z, y)
    const float pxv[3] = {xc0, xc1, xc2};
    const float pyv[3] = {xc2, xc0, xc1};

    float* outp = out + (size_t)n * 48;

#pragma unroll
    for (int l = 0; l < NLEV; ++l) {
        const int r = 128 << l;
        const float half_r = 0.5f * (float)r;
        const float* __restrict__ tp = tps[l];

#pragma unroll
        for (int p = 0; p < 3; ++p) {
            // align_corners=False mapping
            const float fx = pxv[p] * half_r + (half_r - 0.5f);
            const float fy = pyv[p] * half_r + (half_r - 0.5f);
            const float fx0 = floorf(fx);
            const float fy0 = floorf(fy);
            const float wx1 = fx - fx0;
            const float wy1 = fy - fy0;
            const int ix0 = (int)fx0;
            const int iy0 = (int)fy0;

            // Validity-masked corner weights (zero padding), clamped indices.
            const float ax0 = (ix0 >= 0)    ? (1.0f - wx1) : 0.0f;
            const float ax1 = (ix0 + 1 < r) ? wx1          : 0.0f;
            const float ay0 = (iy0 >= 0)    ? (1.0f - wy1) : 0.0f;
            const float ay1 = (iy0 + 1 < r) ? wy1          : 0.0f;
            const int cy0 = iy0 < 0 ? 0 : iy0;
            const int cy1 = (iy0 + 1 > r - 1) ? (r - 1) : (iy0 + 1);

            // Adjacent x-pair starting index, and remapped pair weights.
            const int s = ix0 < 0 ? 0 : (ix0 > r - 2 ? r - 2 : ix0);
            float b0 = 0.0f, b1 = 0.0f;
            if (ix0 == s)          { b0 = ax0; b1 = ax1; }  // interior
            else if (ix0 + 1 == s) { b0 = ax1; }            // ix0 == -1 (left edge)
            else if (ix0 == s + 1) { b1 = ax0; }            // ix0 == r-1 (right edge)

            // Precombine bilinear weights once for all 4 channels.
            const float c00 = ay0 * b0, c01 = ay0 * b1;
            const float c10 = ay1 * b0, c11 = ay1 * b1;

            const float* __restrict__ base = tp + (size_t)p * NC * r * r;
            const size_t o0 = (size_t)cy0 * r + s;
            const size_t o1 = (size_t)cy1 * r + s;

            v4f res;
#pragma unroll
            for (int c = 0; c < NC; ++c) {
                const float* __restrict__ gc = base + (size_t)c * r * r;
                v2f q0, q1;  // dword-aligned b64 pair loads
                __builtin_memcpy(&q0, gc + o0, sizeof(v2f));
                __builtin_memcpy(&q1, gc + o1, sizeof(v2f));
                const float v = c00 * q0.x + c01 * q0.y + c10 * q1.x + c11 * q1.y;
                const int e = (l * 3 + p) * 4 + c;
                res[c] = v * (s_m[e] * wl[p]);
            }
            // Stream the 192 MB output with NT stores so the 64 MB of grids stay
            // resident in the 192 MB L2 (the gathers are the L2-bound part).
            __builtin_nontemporal_store(res, (v4f*)(outp + l * 12 + p * 4));
        }
    }
}

extern "C" void kernel_launch(void* const* d_in, const int* in_sizes, int n_in,
                              void* d_out, int out_size, void* d_ws, size_t ws_size,
                              hipStream_t stream) {
    const float* x   = (const float*)d_in[0];
    const float* tp0 = (const float*)d_in[1];
    const float* tp1 = (const float*)d_in[2];
    const float* tp2 = (const float*)d_in[3];
    const float* tp3 = (const float*)d_in[4];
    const float* tl0 = (const float*)d_in[5];
    const float* tl1 = (const float*)d_in[6];
    const float* tl2 = (const float*)d_in[7];
    const float* tl3 = (const float*)d_in[8];
    float* out = (float*)d_out;

    const int N = in_sizes[0] / 3;
    const int blocks = (N + BLK - 1) / BLK;
    triplane_enc_kernel<<<blocks, BLK, 0, stream>>>(x, tp0, tp1, tp2, tp3,
                                                    tl0, tl1, tl2, tl3, out, N);
}